// TitanicGCNDistilled_54451595379032
// MI455X (gfx1250) — compile-verified
//
#include <hip/hip_runtime.h>
#include <hip/hip_bf16.h>

typedef __attribute__((ext_vector_type(2))) float v2f;
typedef __attribute__((ext_vector_type(8))) float v8f;

// ---------------------------------------------------------------------------
// Utility: fill a float buffer with a constant (workspace is poisoned 0xAA).
// ---------------------------------------------------------------------------
__global__ void fill_kernel(float* __restrict__ p, float v, long long n) {
    long long i = (long long)blockIdx.x * blockDim.x + threadIdx.x;
    if (i < n) p[i] = v;
}

// ---------------------------------------------------------------------------
// Degree count from edge dst (self-loops pre-seeded via fill with 1.0).
// ---------------------------------------------------------------------------
__global__ void degree_kernel(const long long* __restrict__ dst,
                              float* __restrict__ deg, int E) {
    int e = blockIdx.x * blockDim.x + threadIdx.x;
    if (e < E) atomicAdd(&deg[(int)dst[e]], 1.0f);
}

// deg -> deg^{-1/2} in place (deg >= 1 always due to self-loop)
__global__ void dinv_kernel(float* __restrict__ deg, int n) {
    int i = blockIdx.x * blockDim.x + threadIdx.x;
    if (i < n) deg[i] = rsqrtf(deg[i]);
}

// ---------------------------------------------------------------------------
// WMMA fp32 GEMM: C[M,N] = A[M,K] @ B[K,N], row-major, M % 16 == 0,
// N % 16 == 0, K % 4 == 0. One wave (32 lanes) computes one 16x16 tile
// using V_WMMA_F32_16X16X4_F32 across the K loop.
// A 16x4 layout: lane L holds row M=L%16; VGPR0/1 = K = (L/16)*2 + {0,1}.
// B 4x16 layout: lane L holds col N=L%16; VGPR0/1 = K = (L/16)*2 + {0,1}.
// C layout: VGPR v: lanes 0-15 -> (M=v, N=lane); lanes 16-31 -> (M=v+8).
// ---------------------------------------------------------------------------
__global__ void wmma_gemm_f32(const float* __restrict__ A,
                              const float* __restrict__ B,
                              float* __restrict__ C,
                              int N, int K) {
    const int lane = threadIdx.x;        // 0..31 (wave32)
    const int half = lane >> 4;          // 0 or 1
    const int idx  = lane & 15;
    const int row0 = blockIdx.x * 16;
    const int col0 = blockIdx.y * 16;

    const float* Arow = A + (size_t)(row0 + idx) * K;
    v8f acc = {};

    for (int k = 0; k < K; k += 4) {
        const int ka = k + half * 2;
        v2f a, b;
        a.x = Arow[ka];
        a.y = Arow[ka + 1];
        const float* Bp = B + (size_t)ka * N + col0 + idx;
        b.x = Bp[0];
        b.y = Bp[N];
        // (neg_a, A, neg_b, B, c_mod, C, reuse_a, reuse_b)
        acc = __builtin_amdgcn_wmma_f32_16x16x4_f32(
            false, a, false, b, (short)0, acc, false, false);
    }

    const int mbase = row0 + half * 8;
    float* Cp = C + (size_t)mbase * N + col0 + idx;
#pragma unroll
    for (int v = 0; v < 8; ++v) Cp[(size_t)v * N] = acc[v];
}

// ---------------------------------------------------------------------------
// Edge aggregation: for each edge (incl. virtual self-loops e >= E),
// agg[dst] += h[src] * dinv[src]*dinv[dst].  grid.y indexes float4 chunks
// of the feature dimension.  Working set fits in the 192MB L2, so the
// random gather + atomic scatter stays on-chip.
// ---------------------------------------------------------------------------
__global__ void aggregate_kernel(const float* __restrict__ h,
                                 const long long* __restrict__ src,
                                 const long long* __restrict__ dst,
                                 const float* __restrict__ dinv,
                                 float* __restrict__ agg,
                                 int E, int Etot, int F) {
    int e = blockIdx.x * blockDim.x + threadIdx.x;
    if (e >= Etot) return;
    int s, d;
    if (e < E) { s = (int)src[e]; d = (int)dst[e]; }
    else       { s = d = e - E; }                    // self-loop
    const float w = dinv[s] * dinv[d];
    const int c = blockIdx.y * 4;
    const float4 hv = *(const float4*)(h + (size_t)s * F + c);
    float* ap = agg + (size_t)d * F + c;
    atomicAdd(ap + 0, hv.x * w);
    atomicAdd(ap + 1, hv.y * w);
    atomicAdd(ap + 2, hv.z * w);
    atomicAdd(ap + 3, hv.w * w);
}

// y = relu(y + bias[f]) in place. F is a power of two (64 / 32).
__global__ void bias_relu_kernel(float* __restrict__ h,
                                 const float* __restrict__ b,
                                 int Fmask, long long total) {
    long long i = (long long)blockIdx.x * blockDim.x + threadIdx.x;
    if (i >= total) return;
    float v = h[i] + b[(int)(i & Fmask)];
    h[i] = v > 0.0f ? v : 0.0f;
}

// Output head: out[i,:] = h3[i,:] @ out_w (32x2) + out_b
__global__ void out_kernel(const float* __restrict__ h3,
                           const float* __restrict__ ow,
                           const float* __restrict__ ob,
                           float* __restrict__ out, int n) {
    int i = blockIdx.x * blockDim.x + threadIdx.x;
    if (i >= n) return;
    const float* r = h3 + (size_t)i * 32;
    float o0 = ob[0], o1 = ob[1];
#pragma unroll
    for (int k = 0; k < 32; ++k) {
        float hv = r[k];
        o0 += hv * ow[k * 2 + 0];
        o1 += hv * ow[k * 2 + 1];
    }
    out[(size_t)i * 2 + 0] = o0;
    out[(size_t)i * 2 + 1] = o1;
}

// ---------------------------------------------------------------------------
// Launch
// ---------------------------------------------------------------------------
extern "C" void kernel_launch(void* const* d_in, const int* in_sizes, int n_in,
                              void* d_out, int out_size, void* d_ws, size_t ws_size,
                              hipStream_t stream) {
    const float*     x    = (const float*)d_in[0];
    const long long* ei   = (const long long*)d_in[1];  // int64 per reference
    const float*     W1   = (const float*)d_in[2];
    const float*     b1   = (const float*)d_in[3];
    const float*     W2   = (const float*)d_in[4];
    const float*     b2   = (const float*)d_in[5];
    const float*     fc_w = (const float*)d_in[6];
    const float*     fc_b = (const float*)d_in[7];
    const float*     ow   = (const float*)d_in[8];
    const float*     obv  = (const float*)d_in[9];

    const int hid  = in_sizes[3];             // 64
    const int hid2 = in_sizes[5];             // 32
    const int in_c = in_sizes[2] / hid;       // 128
    const int N    = in_sizes[0] / in_c;      // 100000 (multiple of 16)
    const int E    = in_sizes[1] / 2;         // 3,200,000
    const int Etot = E + N;

    const long long* src = ei;
    const long long* dst = ei + E;

    // Workspace layout (floats): deg | bufA(N*hid) | bufB(N*hid) | bufC(N*hid2) | bufD(N*hid2)
    float* deg  = (float*)d_ws;
    float* bufA = deg  + N;
    float* bufB = bufA + (size_t)N * hid;
    float* bufC = bufB + (size_t)N * hid;
    float* bufD = bufC + (size_t)N * hid2;

    const int TB = 256;
    auto blocks = [](long long n, int tb) { return (unsigned)((n + tb - 1) / tb); };

    // --- degree / dinv ---
    fill_kernel<<<blocks(N, TB), TB, 0, stream>>>(deg, 1.0f, N);  // self-loops
    degree_kernel<<<blocks(E, TB), TB, 0, stream>>>(dst, deg, E);
    dinv_kernel<<<blocks(N, TB), TB, 0, stream>>>(deg, N);

    // --- layer 1: h1 = x @ W1 ---
    wmma_gemm_f32<<<dim3(N / 16, hid / 16), 32, 0, stream>>>(x, W1, bufA, hid, in_c);
    fill_kernel<<<blocks((long long)N * hid, TB), TB, 0, stream>>>(bufB, 0.0f, (long long)N * hid);
    aggregate_kernel<<<dim3(blocks(Etot, TB), hid / 4), TB, 0, stream>>>(
        bufA, src, dst, deg, bufB, E, Etot, hid);
    bias_relu_kernel<<<blocks((long long)N * hid, TB), TB, 0, stream>>>(
        bufB, b1, hid - 1, (long long)N * hid);

    // --- layer 2: h2 = h1a @ W2 ---
    wmma_gemm_f32<<<dim3(N / 16, hid2 / 16), 32, 0, stream>>>(bufB, W2, bufC, hid2, hid);
    fill_kernel<<<blocks((long long)N * hid2, TB), TB, 0, stream>>>(bufD, 0.0f, (long long)N * hid2);
    aggregate_kernel<<<dim3(blocks(Etot, TB), hid2 / 4), TB, 0, stream>>>(
        bufC, src, dst, deg, bufD, E, Etot, hid2);
    bias_relu_kernel<<<blocks((long long)N * hid2, TB), TB, 0, stream>>>(
        bufD, b2, hid2 - 1, (long long)N * hid2);

    // --- FC: h3 = relu(h2a @ fc_w + fc_b)  (overwrite bufC; gemm reads bufD) ---
    wmma_gemm_f32<<<dim3(N / 16, hid2 / 16), 32, 0, stream>>>(bufD, fc_w, bufC, hid2, hid2);
    bias_relu_kernel<<<blocks((long long)N * hid2, TB), TB, 0, stream>>>(
        bufC, fc_b, hid2 - 1, (long long)N * hid2);

    // --- output head ---
    out_kernel<<<blocks(N, TB), TB, 0, stream>>>(bufC, ow, obv, (float*)d_out, N);
}